// TermLevelLoss_24696061952429
// MI455X (gfx1250) — compile-verified
//
#include <hip/hip_runtime.h>
#include <hip/hip_bf16.h>
#include <math.h>

// Problem constants (from the reference).
#define B_SZ   256
#define V_SZ   128000
#define L_KO   32
#define L_EN   32
#define N_NEG  1024
#define EPS_F       1e-8f
#define MARGIN_F    2.0f
#define NEG_THRESH  0.1f

typedef __attribute__((ext_vector_type(2))) float v2f;
typedef __attribute__((ext_vector_type(8))) float v8f;

// Deterministic wave32-wide sum via the fp32 matrix pipe (confirmed lowering:
// v_wmma_f32_16x16x4_f32). Lane L supplies A[L%16][L<16?0:2] = x (other K
// slot 0), B = 4x16 all-ones, so D[m][*] = x_m + x_{m+16}. Lane n holds rows
// 0..7 of D, lane n+16 rows 8..15; one xor-16 shuffle completes the 32-lane
// sum in exact fp32.
__device__ __forceinline__ float wave_sum_f32(float x) {
#if defined(__gfx1250__) && __has_builtin(__builtin_amdgcn_wmma_f32_16x16x4_f32)
  v2f a; a[0] = x;    a[1] = 0.0f;
  v2f b; b[0] = 1.0f; b[1] = 1.0f;
  v8f c = {};
  c = __builtin_amdgcn_wmma_f32_16x16x4_f32(
      /*neg_a=*/false, a, /*neg_b=*/false, b,
      /*c_mod=*/(short)0, c, /*reuse_a=*/false, /*reuse_b=*/false);
  float s = ((c[0] + c[1]) + (c[2] + c[3])) + ((c[4] + c[5]) + (c[6] + c[7]));
  s += __shfl_xor(s, 16, 32);
  return s;
#else
  for (int off = 16; off > 0; off >>= 1) x += __shfl_xor(x, off, 32);
  return x;
#endif
}

// Kernel 1: per-batch-row negative-loss partial sums.
// Block b handles row b. Thread t loads ids 4t..4t+3 as one int4 (coalesced
// B128) and issues 4 independent gathers -> 4 loads in flight per thread
// instead of a serialized id->gather->wait chain.
__global__ void __launch_bounds__(256)
neg_partial_kernel(const float* __restrict__ rep,
                   const int*   __restrict__ neg_ids,
                   float*       __restrict__ partial) {
  const int b = blockIdx.x;
  const float* row = rep + (size_t)b * V_SZ;

  const int4 ids = ((const int4*)neg_ids)[threadIdx.x];  // 256*4 == N_NEG

  const float v0 = row[ids.x];
  const float v1 = row[ids.y];
  const float v2 = row[ids.z];
  const float v3 = row[ids.w];

  const float r0 = fmaxf(v0 - NEG_THRESH, 0.0f);
  const float r1 = fmaxf(v1 - NEG_THRESH, 0.0f);
  const float r2 = fmaxf(v2 - NEG_THRESH, 0.0f);
  const float r3 = fmaxf(v3 - NEG_THRESH, 0.0f);
  const float acc = (r0 + r1) + (r2 + r3);

  __shared__ float parts[8];
  const float w = wave_sum_f32(acc);          // EXEC all-ones (full block)
  const int lane = threadIdx.x & 31;
  const int wid  = threadIdx.x >> 5;
  if (lane == 0) parts[wid] = w;
  __syncthreads();
  if (threadIdx.x == 0) {
    float t = 0.0f;
    #pragma unroll
    for (int i = 0; i < 8; ++i) t += parts[i];
    partial[b] = t;
  }
}

// Kernel 2 (single block, 256 threads): thread t owns batch row t.
// All 32 ids are always valid, so gather unconditionally (64 outstanding
// loads per thread) and mask the contribution with (i < len) — no divergent
// dependent-load loops. Then deterministically block-reduce the 4 losses.
__global__ void __launch_bounds__(256)
row_loss_kernel(const float* __restrict__ rep,
                const int*   __restrict__ ko_ids,
                const int*   __restrict__ en_ids,
                const int*   __restrict__ ko_len_p,
                const int*   __restrict__ en_len_p,
                const float* __restrict__ neg_partial,
                float*       __restrict__ out) {
  const int t = threadIdx.x;                  // 0..255 == batch row
  const float* row = rep + (size_t)t * V_SZ;

  int kl = ko_len_p[t];
  kl = kl < 0 ? 0 : (kl > L_KO ? L_KO : kl);
  int el = en_len_p[t];
  el = el < 0 ? 0 : (el > L_EN ? L_EN : el);

  // Load all ids up front (8 + 8 coalesced-per-thread B128 loads).
  const int4* kid4 = (const int4*)(ko_ids + t * L_KO);
  const int4* eid4 = (const int4*)(en_ids + t * L_EN);
  int kid[L_KO], eid[L_EN];
  #pragma unroll
  for (int q = 0; q < L_KO / 4; ++q) {
    const int4 a = kid4[q];
    kid[4*q+0] = a.x; kid[4*q+1] = a.y; kid[4*q+2] = a.z; kid[4*q+3] = a.w;
    const int4 b = eid4[q];
    eid[4*q+0] = b.x; eid[4*q+1] = b.y; eid[4*q+2] = b.z; eid[4*q+3] = b.w;
  }

  // Issue all 64 gathers; independent loads stay in flight together.
  float kv[L_KO], ev[L_EN];
  #pragma unroll
  for (int i = 0; i < L_KO; ++i) kv[i] = row[kid[i]];
  #pragma unroll
  for (int i = 0; i < L_EN; ++i) ev[i] = row[eid[i]];

  float ksum = 0.0f, tsum = 0.0f, msum = 0.0f;
  #pragma unroll
  for (int i = 0; i < L_KO; ++i) {
    const float nl = -logf(kv[i] + EPS_F);
    ksum += (i < kl) ? nl : 0.0f;
  }
  #pragma unroll
  for (int i = 0; i < L_EN; ++i) {
    const float nl = -logf(ev[i] + EPS_F);
    const float mg = fmaxf(MARGIN_F - ev[i], 0.0f);
    tsum += (i < el) ? nl : 0.0f;
    msum += (i < el) ? mg : 0.0f;
  }

  float vals[4];
  vals[0] = (kl > 0) ? ksum / (float)kl : 0.0f;   // self_loss row
  vals[1] = (el > 0) ? tsum / (float)el : 0.0f;   // target_loss row
  vals[2] = (el > 0) ? msum / (float)el : 0.0f;   // margin_loss row
  vals[3] = neg_partial[t];                       // negative_loss partial

  __shared__ float parts[4][8];
  const int lane = t & 31;
  const int wid  = t >> 5;
  // Straight-line code above: EXEC is all-ones at each WMMA below.
  #pragma unroll
  for (int k = 0; k < 4; ++k) {
    const float w = wave_sum_f32(vals[k]);
    if (lane == 0) parts[k][wid] = w;
    __syncthreads();
  }
  if (t < 4) {
    float s = 0.0f;
    #pragma unroll
    for (int i = 0; i < 8; ++i) s += parts[t][i];
    out[t] = (t == 3) ? s / (float)(B_SZ * N_NEG) : s / (float)B_SZ;
  }
}

extern "C" void kernel_launch(void* const* d_in, const int* in_sizes, int n_in,
                              void* d_out, int out_size, void* d_ws, size_t ws_size,
                              hipStream_t stream) {
  const float* rep     = (const float*)d_in[0];  // (B, V) f32
  const int*   ko_ids  = (const int*)  d_in[1];  // (B, 32)
  const int*   ko_len  = (const int*)  d_in[2];  // (B,)
  const int*   en_ids  = (const int*)  d_in[3];  // (B, 32)
  const int*   en_len  = (const int*)  d_in[4];  // (B,)
  const int*   neg_ids = (const int*)  d_in[5];  // (1024,)
  float* out     = (float*)d_out;                // 4 scalars
  float* partial = (float*)d_ws;                 // 256 floats of scratch

  neg_partial_kernel<<<B_SZ, 256, 0, stream>>>(rep, neg_ids, partial);
  row_loss_kernel<<<1, 256, 0, stream>>>(rep, ko_ids, en_ids, ko_len, en_len,
                                         partial, out);
}